// RNNBlock_82386062671988
// MI455X (gfx1250) — compile-verified
//
#include <hip/hip_runtime.h>

#define B_SZ 16384
#define T_SZ 79
#define H_SZ 256
#define HP   260   // padded LDS pitch (floats): stride%64 banks == 4 -> conflict-free

typedef __attribute__((ext_vector_type(2))) float v2f;
typedef __attribute__((ext_vector_type(8))) float v8f;

__global__ __launch_bounds__(512, 1)
void rnn_fused_kernel(const float* __restrict__ x0,
                      const float* __restrict__ x1,
                      const float* __restrict__ x2,
                      const float* __restrict__ W_in,
                      const float* __restrict__ U,
                      const float* __restrict__ b_rnn,
                      const float* __restrict__ W_d,
                      const float* __restrict__ b_d,
                      float* __restrict__ out)
{
    __shared__ __align__(16) float hbuf[2][16 * HP];
    __shared__ float xs[3][16];

    const int tid  = threadIdx.x;
    const int wave = tid >> 5;            // 0..15 : output column tile
    const int lane = tid & 31;
    const int l15  = lane & 15;
    const int half = lane >> 4;           // 0 or 1
    const int col  = (wave << 4) + l15;   // 0..255
    const int r0   = blockIdx.x * 16;     // first batch row of this block

    // zero-init both h buffers (h0 = 0)
    for (int i = tid; i < 2 * 16 * HP; i += 512) ((float*)hbuf)[i] = 0.f;

    // per-lane weights for the xt = x @ W_in + b_rnn term (col fixed per lane)
    const float w0 = W_in[0 * H_SZ + col];
    const float w1 = W_in[1 * H_SZ + col];
    const float w2 = W_in[2 * H_SZ + col];
    const float bb = b_rnn[col];

    // Load this wave's B-operand tiles of U into registers (WMMA B layout:
    // lane n<16 holds U[4k+0..1, col], lane n+16 holds U[4k+2..3, col])
    v2f Ureg[64];
#pragma unroll
    for (int kt = 0; kt < 64; ++kt) {
        const int krow = (kt << 2) + (half << 1);
        v2f b;
        b.x = U[krow * H_SZ + col];
        b.y = U[(krow + 1) * H_SZ + col];
        Ureg[kt] = b;
    }

    // A-operand per-lane base byte offset inside one h buffer:
    // lane m (0..15): row m, K=4k+0..1 ; lane m+16: row m, K=4k+2..3
    const int a_off = l15 * (HP * 4) + half * 8;

    __syncthreads();   // h zero-init visible

    int cur = 0;
    for (int s = 0; s < T_SZ; ++s) {
        const int idx = (T_SZ - 1) - s;   // time reversal
        if (tid < 48) {                   // stage x0/x1/x2 for this block's 16 rows
            const int f = tid >> 4, m = tid & 15;
            const float* xp = (f == 0) ? x0 : ((f == 1) ? x1 : x2);
            xs[f][m] = xp[(r0 + m) * T_SZ + idx];
        }
        __syncthreads();

        v8f acc = {0.f, 0.f, 0.f, 0.f, 0.f, 0.f, 0.f, 0.f};
        const char* abase = (const char*)(&hbuf[cur][0]) + a_off;
#pragma unroll
        for (int kt = 0; kt < 64; ++kt) {
            v2f a = *(const v2f*)(abase + kt * 16);
            acc = __builtin_amdgcn_wmma_f32_16x16x4_f32(
                      false, a, false, Ureg[kt], (short)0, acc, false, false);
        }

        // h_new = relu(acc + xt); C layout: reg v -> row (v + 8*half), col fixed
        float* hn = &hbuf[cur ^ 1][0];
#pragma unroll
        for (int v = 0; v < 8; ++v) {
            const int m = v + (half << 3);
            const float xv = xs[0][m] * w0 + xs[1][m] * w1 + xs[2][m] * w2 + bb;
            float hv = acc[v] + xv;
            hv = fmaxf(hv, 0.f);
            hn[m * HP + col] = hv;
        }
        __syncthreads();
        cur ^= 1;
    }

    // out = hT @ W_d + b_d  (same tiling; W_d B-tiles streamed from L2)
    {
        v8f acc = {0.f, 0.f, 0.f, 0.f, 0.f, 0.f, 0.f, 0.f};
        const char* abase = (const char*)(&hbuf[cur][0]) + a_off;
#pragma unroll
        for (int kt = 0; kt < 64; ++kt) {
            v2f a = *(const v2f*)(abase + kt * 16);
            const int krow = (kt << 2) + (half << 1);
            v2f b;
            b.x = W_d[krow * H_SZ + col];
            b.y = W_d[(krow + 1) * H_SZ + col];
            acc = __builtin_amdgcn_wmma_f32_16x16x4_f32(
                      false, a, false, b, (short)0, acc, false, false);
        }
        const float bd = b_d[col];
#pragma unroll
        for (int v = 0; v < 8; ++v) {
            const int m = v + (half << 3);
            out[(r0 + m) * H_SZ + col] = acc[v] + bd;
        }
    }
}

extern "C" void kernel_launch(void* const* d_in, const int* in_sizes, int n_in,
                              void* d_out, int out_size, void* d_ws, size_t ws_size,
                              hipStream_t stream) {
    (void)in_sizes; (void)n_in; (void)d_ws; (void)ws_size; (void)out_size;
    const float* x0    = (const float*)d_in[0];
    const float* x1    = (const float*)d_in[1];
    const float* x2    = (const float*)d_in[2];
    const float* W_in  = (const float*)d_in[3];
    const float* U     = (const float*)d_in[4];
    const float* b_rnn = (const float*)d_in[5];
    const float* W_d   = (const float*)d_in[6];
    const float* b_d   = (const float*)d_in[7];
    float* out = (float*)d_out;

    dim3 grid(B_SZ / 16), block(512);
    rnn_fused_kernel<<<grid, block, 0, stream>>>(x0, x1, x2, W_in, U, b_rnn,
                                                 W_d, b_d, out);
}